// PointConvNet_83854941487655
// MI455X (gfx1250) — compile-verified
//
#include <hip/hip_runtime.h>

typedef float v2f __attribute__((ext_vector_type(2)));
typedef float v8f __attribute__((ext_vector_type(8)));

#define B_CLOUDS 16
#define N_PER    2048
#define NPTS     (B_CLOUDS * N_PER)      // 32768
#define MAX_NB   32
#define E_PAD    (NPTS * (MAX_NB + 1))   // 1081344
#define HID      64
#define OUTF     64
#define TILES    (E_PAD / 16)            // 67584
#define WPB      8                       // waves per block
#define BLOCK    (WPB * 32)

// ---------------------------------------------------------------------------
// Init: out[0 .. N*64) = -inf (segment-max identity), then append pos copy
// and batch ids (tuple outputs concatenated flat).
// ---------------------------------------------------------------------------
__global__ void pc_init(float* __restrict__ out, const float* __restrict__ pos,
                        int out_size) {
  int i = blockIdx.x * blockDim.x + threadIdx.x;
  const float NEG_INF = -__builtin_inff();
  if (i < NPTS * OUTF && i < out_size) out[i] = NEG_INF;
  int p = NPTS * OUTF + i;
  if (i < NPTS * 3 && p < out_size) out[p] = pos[i];
  int q = NPTS * OUTF + NPTS * 3 + i;
  if (i < NPTS && q < out_size) out[q] = (float)(i / N_PER);
}

// ---------------------------------------------------------------------------
// Main kernel: one wave == one 16-edge tile.
//   msg(16x8, padded K) @ W1(8x64) -> +b1 -> relu -> h(16x64)
//   h(16x64) @ W2(64x64) -> +b2 -> atomic max scatter to out[dst]
// All matmuls via v_wmma_f32_16x16x4_f32.
// ---------------------------------------------------------------------------
__global__ __launch_bounds__(BLOCK) void pc_wmma(
    const float* __restrict__ x, const float* __restrict__ pos,
    const long long* __restrict__ src, const long long* __restrict__ dst,
    const float* __restrict__ W1, const float* __restrict__ b1,
    const float* __restrict__ W2, const float* __restrict__ b2,
    float* __restrict__ out) {
  __shared__ float W1s[8 * 64];          // K padded 7 -> 8 (row 7 = 0)
  __shared__ float b1s[64];
  __shared__ float W2s[64 * 64];
  __shared__ float b2s[64];
  __shared__ float msgS[WPB * 16 * 9];   // per-wave message tile, stride 9
  __shared__ int   dstS[WPB * 16];
  __shared__ float hS[WPB * 16 * 68];    // per-wave hidden tile, stride 68

  const int tid = threadIdx.x;

  // Cooperative weight staging
  for (int i = tid; i < 8 * 64; i += BLOCK) {
    int r = i >> 6, c = i & 63;
    W1s[i] = (r < 7) ? W1[r * 64 + c] : 0.0f;
  }
  for (int i = tid; i < 64; i += BLOCK) { b1s[i] = b1[i]; b2s[i] = b2[i]; }
  for (int i = tid; i < 64 * 64; i += BLOCK) W2s[i] = W2[i];
  __syncthreads();

  const int wave = tid >> 5;
  const int lane = tid & 31;
  const int tile = blockIdx.x * WPB + wave;
  if (tile >= TILES) return;

  float* msg = &msgS[wave * 16 * 9];
  int*   dI  = &dstS[wave * 16];
  float* hT  = &hS[wave * 16 * 68];

  // ---- gather 16 edges (lanes 0..15) --------------------------------------
  if (lane < 16) {
    long long e = (long long)tile * 16 + lane;
    long long s = src[e];
    long long d = dst[e];
    dI[lane] = (int)d;
    const float* xs = x + s * 4;
    const float* ps = pos + s * 3;
    const float* pd = pos + d * 3;
    float* m = msg + lane * 9;
    m[0] = xs[0]; m[1] = xs[1]; m[2] = xs[2]; m[3] = xs[3];
    m[4] = ps[0] - pd[0];
    m[5] = ps[1] - pd[1];
    m[6] = ps[2] - pd[2];
    m[7] = 0.0f;  m[8] = 0.0f;
  }
  asm volatile("s_wait_dscnt 0" ::: "memory");

  const int row  = lane & 15;          // A-matrix row (M)
  const int col  = lane & 15;          // B/C column within N-tile
  const int koff = (lane >> 4) << 1;   // lanes 0-15 -> K{0,1}; 16-31 -> K{2,3}
  const int half = (lane >> 4) << 3;   // C rows: +8 for upper lanes

  // ---- GEMM1: msg(16x8) @ W1(8x64), bias in accumulator -------------------
  v8f acc[4];
#pragma unroll
  for (int n = 0; n < 4; ++n) {
    float bv = b1s[n * 16 + col];
    acc[n] = (v8f){bv, bv, bv, bv, bv, bv, bv, bv};
  }
#pragma unroll
  for (int s = 0; s < 2; ++s) {
    v2f a;
    a.x = msg[row * 9 + s * 4 + koff];
    a.y = msg[row * 9 + s * 4 + koff + 1];
#pragma unroll
    for (int n = 0; n < 4; ++n) {
      v2f b;
      b.x = W1s[(s * 4 + koff) * 64 + n * 16 + col];
      b.y = W1s[(s * 4 + koff + 1) * 64 + n * 16 + col];
      acc[n] = __builtin_amdgcn_wmma_f32_16x16x4_f32(
          false, a, false, b, (short)0, acc[n], false, false);
    }
  }

  // ---- ReLU, stage h into LDS (stride 68) ---------------------------------
#pragma unroll
  for (int n = 0; n < 4; ++n) {
#pragma unroll
    for (int g = 0; g < 8; ++g) {
      float v = acc[n][g];
      v = v > 0.0f ? v : 0.0f;
      hT[(g + half) * 68 + n * 16 + col] = v;
    }
  }
  asm volatile("s_wait_dscnt 0" ::: "memory");

  // ---- GEMM2: h(16x64) @ W2(64x64), bias in accumulator -------------------
  v8f acc2[4];
#pragma unroll
  for (int n = 0; n < 4; ++n) {
    float bv = b2s[n * 16 + col];
    acc2[n] = (v8f){bv, bv, bv, bv, bv, bv, bv, bv};
  }
#pragma unroll
  for (int k = 0; k < 16; ++k) {
    v2f a;
    a.x = hT[row * 68 + k * 4 + koff];
    a.y = hT[row * 68 + k * 4 + koff + 1];
#pragma unroll
    for (int n = 0; n < 4; ++n) {
      v2f b;
      b.x = W2s[(k * 4 + koff) * 64 + n * 16 + col];
      b.y = W2s[(k * 4 + koff + 1) * 64 + n * 16 + col];
      acc2[n] = __builtin_amdgcn_wmma_f32_16x16x4_f32(
          false, a, false, b, (short)0, acc2[n], false, false);
    }
  }

  // ---- segment-max scatter via native float-max atomics -------------------
#pragma unroll
  for (int g = 0; g < 8; ++g) {
    int m = g + half;
    long long d = (long long)dI[m];
    float* basep = out + d * OUTF;
#pragma unroll
    for (int n = 0; n < 4; ++n) {
      float v = acc2[n][g];
      float* p = basep + n * 16 + col;
      asm volatile("global_atomic_max_num_f32 %0, %1, off"
                   :: "v"(p), "v"(v) : "memory");
    }
  }
}

// ---------------------------------------------------------------------------
extern "C" void kernel_launch(void* const* d_in, const int* in_sizes, int n_in,
                              void* d_out, int out_size, void* d_ws,
                              size_t ws_size, hipStream_t stream) {
  const float*     x    = (const float*)d_in[0];
  const float*     pos  = (const float*)d_in[1];
  // d_in[2] = batch (recomputed analytically in pc_init)
  const long long* eidx = (const long long*)d_in[3];
  const float*     W1   = (const float*)d_in[4];
  const float*     b1   = (const float*)d_in[5];
  const float*     W2   = (const float*)d_in[6];
  const float*     b2   = (const float*)d_in[7];
  float*           out  = (float*)d_out;

  const long long* src = eidx;
  const long long* dst = eidx + E_PAD;

  int initThreads = NPTS * OUTF;
  pc_init<<<(initThreads + 255) / 256, 256, 0, stream>>>(out, pos, out_size);

  pc_wmma<<<TILES / WPB, BLOCK, 0, stream>>>(x, pos, src, dst, W1, b1, W2, b2,
                                             out);
}